// DebugChamferLoss_5085241278567
// MI455X (gfx1250) — compile-verified
//
#include <hip/hip_runtime.h>

// CDNA5 / gfx1250 chamfer NN-distance using V_WMMA_F32_16X16X4_F32.
// d2(m,n) = |s_m|^2 + |t_n|^2 - 2 s.t  ; K=3 padded to K=4:
//   A row m = (sx, sy, sz, 1), B col n = (-2tx, -2ty, -2tz, tsq)
//   => C = -2*dot + tsq ; |s|^2 add and the >=0 clamp are hoisted out of the
//   sweep (both commute with the min reduction).
// B-tile LDS reads: the half-dependent slot select is folded into the LDS
// address (float2 at (t+sub)*16 + half*8), so the hot loop has no cndmasks:
//   ds_load_b64 -> v_wmma_f32_16x16x4_f32 -> 8x v_min (fused min3).

typedef __attribute__((ext_vector_type(2))) float v2f;
typedef __attribute__((ext_vector_type(8))) float v8f;

#define WAVE            32
#define WAVES_PER_BLOCK 8
#define BLOCK           (WAVE * WAVES_PER_BLOCK)   // 256 threads = 8 wave32
#define ROWS_PER_BLOCK  (WAVES_PER_BLOCK * 16)     // 128 source rows / block
#define CHUNK           2048                       // targets staged in LDS (32 KB)

__global__ __launch_bounds__(BLOCK)
void chamfer_nn_kernel(const float* __restrict__ src, const float* __restrict__ smask,
                       const float* __restrict__ tgt, const float* __restrict__ tmask,
                       int Ns, int Nt, float* __restrict__ out)
{
    __shared__ float4 tile[CHUNK];   // per target: {(-2x,-2y), (-2z,tsq)} as 2x float2

    const int b    = blockIdx.y;
    const int lane = threadIdx.x & (WAVE - 1);
    const int wid  = threadIdx.x >> 5;
    const int half = lane >> 4;      // 0: K0/K1 slots, 1: K2/K3 slots
    const int sub  = lane & 15;      // row (A) / column (B) index in tile

    // ---- build this wave's A tile: 16 source rows, held for the whole sweep
    const int    row0 = blockIdx.x * ROWS_PER_BLOCK + wid * 16;
    const int    srow = row0 + sub;
    const float* sp   = src + ((size_t)b * Ns + srow) * 3;
    float sx = sp[0], sy = sp[1], sz = sp[2];
    if (smask) { float m = smask[(size_t)b * Ns + srow]; sx *= m; sy *= m; sz *= m; }
    const float ssq = sx * sx + sy * sy + sz * sz;

    v2f a;                            // lane m: (K0,K1)=(x,y); lane m+16: (K2,K3)=(z,1)
    a.x = half ? sz   : sx;
    a.y = half ? 1.0f : sy;

    // per-lane B read base: column sub, half-dependent float2 slot
    const float2* tile2 = (const float2*)tile;         // tile2[2*t + s]
    const int     bidx0 = 2 * sub + half;              // + 2*t per tile step

    float rmin[8];
#pragma unroll
    for (int v = 0; v < 8; ++v) rmin[v] = 3.0e38f;

    for (int cbase = 0; cbase < Nt; cbase += CHUNK) {
        // ---- cooperative stage of CHUNK targets into LDS
#pragma unroll
        for (int i = 0; i < CHUNK / BLOCK; ++i) {
            const int    t  = i * BLOCK + (int)threadIdx.x;
            const int    g  = cbase + t;
            const float* tp = tgt + ((size_t)b * Nt + g) * 3;
            float tx = tp[0], ty = tp[1], tz = tp[2];
            if (tmask) { float m = tmask[(size_t)b * Nt + g]; tx *= m; ty *= m; tz *= m; }
            const float tsq = tx * tx + ty * ty + tz * tz;
            tile[t] = make_float4(-2.0f * tx, -2.0f * ty, -2.0f * tz, tsq);
        }
        __syncthreads();

        // ---- sweep: one 16x16 tile per iteration via V_WMMA_F32_16X16X4_F32
        for (int t = 0; t < CHUNK; t += 16) {
            const float2 p = tile2[2 * t + bidx0];     // ds_load_b64, conflict-free
            v2f bb;
            bb.x = p.x;
            bb.y = p.y;
            v8f c = {};
            c = __builtin_amdgcn_wmma_f32_16x16x4_f32(
                    false, a, false, bb, (short)0, c, false, false);
#pragma unroll
            for (int v = 0; v < 8; ++v)
                rmin[v] = fminf(rmin[v], c[v]);        // ssq add + clamp deferred
        }
        __syncthreads();
    }

    // ---- fold row-mins across the 16 lanes of each half
#pragma unroll
    for (int v = 0; v < 8; ++v) {
#pragma unroll
        for (int off = 1; off < 16; off <<= 1)
            rmin[v] = fminf(rmin[v], __shfl_xor(rmin[v], off, WAVE));
    }

    // per-row |s|^2 lives in lane (row - row0) of either half; fetch via shuffle
    float ssq_row[8];
#pragma unroll
    for (int v = 0; v < 8; ++v)
        ssq_row[v] = __shfl(ssq, half * 8 + v, WAVE);

    if (sub == 0) {                  // lanes 0 & 16 each write 8 contiguous rows
        float r[8];
#pragma unroll
        for (int v = 0; v < 8; ++v)
            r[v] = sqrtf(fmaxf(rmin[v] + ssq_row[v], 0.0f)) * 100.0f;
        float* op = out + (size_t)b * Ns + row0 + half * 8;   // 32B aligned
        ((float4*)op)[0] = make_float4(r[0], r[1], r[2], r[3]);
        ((float4*)op)[1] = make_float4(r[4], r[5], r[6], r[7]);
    }
}

// out1 holds raw pred->gt distance; produce conf-weighted loss and masked loss.
__global__ void chamfer_epilogue(const float* __restrict__ conf,
                                 const float* __restrict__ mask,
                                 float* __restrict__ out0,
                                 float* __restrict__ out1, int n)
{
    const int i = blockIdx.x * blockDim.x + threadIdx.x;
    if (i < n) {
        const float raw = out1[i];
        const float c   = conf[i];
        const float m   = mask[i];
        out0[i] = (raw * c - logf(c)) * m;   // ALPHA = 1.0
        out1[i] = raw * m;
    }
}

extern "C" void kernel_launch(void* const* d_in, const int* in_sizes, int n_in,
                              void* d_out, int out_size, void* d_ws, size_t ws_size,
                              hipStream_t stream)
{
    const float* x_gt   = (const float*)d_in[0];  // (B, V1, 3)
    const float* x_pred = (const float*)d_in[1];  // (B, V2, 3)
    const float* mask   = (const float*)d_in[2];  // (B, V2)
    const float* conf   = (const float*)d_in[3];  // (B, V2)

    const int B  = 2;
    const int V2 = in_sizes[2] / B;               // 8192
    const int V1 = in_sizes[0] / (B * 3);         // 8192

    float* out0 = (float*)d_out;                  // loss_pred2gt_conf (B*V2)
    float* out1 = out0 + (size_t)B * V2;          // loss_pred2gt      (B*V2)
    float* out2 = out1 + (size_t)B * V2;          // loss_gt2pred      (B*V1)

    // pred -> gt : source = masked pred, target = gt (raw distances into out1)
    dim3 g1(V2 / ROWS_PER_BLOCK, B);
    chamfer_nn_kernel<<<g1, BLOCK, 0, stream>>>(x_pred, mask, x_gt, nullptr, V2, V1, out1);

    // gt -> pred : source = gt, target = masked pred
    dim3 g2(V1 / ROWS_PER_BLOCK, B);
    chamfer_nn_kernel<<<g2, BLOCK, 0, stream>>>(x_gt, nullptr, x_pred, mask, V1, V2, out2);

    // confidence weighting + masking (reads raw out1, rewrites it masked)
    const int n = B * V2;
    chamfer_epilogue<<<(n + 255) / 256, 256, 0, stream>>>(conf, mask, out0, out1, n);
}